// IouLoss_2D_3D_37220186587938
// MI455X (gfx1250) — compile-verified
//
#include <hip/hip_runtime.h>
#include <math.h>

#define IOU_EPS 1e-8f

typedef __attribute__((ext_vector_type(2))) float v2f;
typedef __attribute__((ext_vector_type(8))) float v8f;

// ---------------------------------------------------------------------------
// Wave32 sum-reduction using the CDNA5 WMMA unit.
// A (16x4 f32) lives in 2 VGPRs; we put each lane's partial in VGPR0 and zero
// in VGPR1. With B = ones(4x16):  D[m][n] = sum_k A[m][k], i.e. every D entry
// in row m holds rowsum_m = partial[m] + partial[m+16].  Per the documented
// C/D layout (VGPR k: lanes 0-15 -> M=k, lanes 16-31 -> M=k+8), summing the 8
// D VGPRs per lane yields rows 0-7 (lanes 0-15) or rows 8-15 (lanes 16-31);
// one xor-16 shuffle then gives the full 32-lane total in every lane.
// ---------------------------------------------------------------------------
__device__ __forceinline__ float wave_reduce_wmma(float v) {
    v2f a; a.x = v;    a.y = 0.0f;
    v2f b; b.x = 1.0f; b.y = 1.0f;
    v8f c = {};
    c = __builtin_amdgcn_wmma_f32_16x16x4_f32(
            /*neg_a=*/false, a, /*neg_b=*/false, b,
            /*c_mod=*/(short)0, c, /*reuse_a=*/false, /*reuse_b=*/false);
    float s = c.s0 + c.s1 + c.s2 + c.s3 + c.s4 + c.s5 + c.s6 + c.s7;
    s += __shfl_xor(s, 16, 32);
    return s;
}

// Block-wide sum; every thread returns the block total. lds must hold >= 8 floats.
__device__ __forceinline__ float block_reduce(float v, float* lds) {
    float w = wave_reduce_wmma(v);
    const int lid = threadIdx.x & 31;
    const int wid = threadIdx.x >> 5;
    __syncthreads();                 // protect lds across consecutive calls
    if (lid == 0) lds[wid] = w;
    __syncthreads();
    const int nw = blockDim.x >> 5;
    float r = (lid < nw) ? lds[lid] : 0.0f;
    return wave_reduce_wmma(r);
}

// ---------------------------------------------------------------------------
// Kernels
// ---------------------------------------------------------------------------

// Zero the volume with 128-bit stores (grid-stride over float4s).
__global__ void k_zero(float4* __restrict__ vol4, int n4) {
    const int stride = gridDim.x * blockDim.x;
    const float4 z = make_float4(0.f, 0.f, 0.f, 0.f);
    for (int i = blockIdx.x * blockDim.x + threadIdx.x; i < n4; i += stride)
        vol4[i] = z;
}

// Gather from output_2D via index, scatter into the volume at midxyz.
__global__ void k_scatter(const float* __restrict__ o2d,
                          const int*   __restrict__ index,
                          const int*   __restrict__ midxyz,
                          float*       __restrict__ vol,
                          int n, int D) {
    const int i = blockIdx.x * blockDim.x + threadIdx.x;
    if (i >= n) return;
    const float v = o2d[index[i]];
    const int x = midxyz[3 * i + 0];
    const int y = midxyz[3 * i + 1];
    const int z = midxyz[3 * i + 2];
    vol[((size_t)x * D + y) * D + z] = v;
}

// Streaming IoU partial sums over a pair of arrays (float4 loads):
// per block writes {sum(a*b), sum(a), sum(b)}.
__global__ void k_reduce(const float4* __restrict__ a4,
                         const float4* __restrict__ b4,
                         int n4,
                         float* __restrict__ partials) {
    float si = 0.f, sa = 0.f, sb = 0.f;
    const int stride = gridDim.x * blockDim.x;
    for (int i = blockIdx.x * blockDim.x + threadIdx.x; i < n4; i += stride) {
        const float4 av = a4[i];
        const float4 bv = b4[i];
        si += av.x * bv.x + av.y * bv.y + av.z * bv.z + av.w * bv.w;
        sa += av.x + av.y + av.z + av.w;
        sb += bv.x + bv.y + bv.z + bv.w;
    }
    __shared__ float lds[8];
    const float ti = block_reduce(si, lds);
    const float ta = block_reduce(sa, lds);
    const float tb = block_reduce(sb, lds);
    if (threadIdx.x == 0) {
        partials[3 * blockIdx.x + 0] = ti;
        partials[3 * blockIdx.x + 1] = ta;
        partials[3 * blockIdx.x + 2] = tb;
    }
}

// Single-block final reduction (fixed order -> deterministic): combine block
// partials for both IoUs and emit (loss1, loss2, loss).
__global__ void k_final(const float* __restrict__ partials,
                        int nb2, int nb3,
                        float* __restrict__ out) {
    float i1 = 0.f, o1 = 0.f, m1 = 0.f;
    float i2 = 0.f, v2 = 0.f, m2 = 0.f;
    for (int b = threadIdx.x; b < nb2; b += blockDim.x) {
        i1 += partials[3 * b + 0];
        o1 += partials[3 * b + 1];
        m1 += partials[3 * b + 2];
    }
    const float* p3 = partials + 3 * (size_t)nb2;
    for (int b = threadIdx.x; b < nb3; b += blockDim.x) {
        i2 += p3[3 * b + 0];
        v2 += p3[3 * b + 1];
        m2 += p3[3 * b + 2];
    }
    __shared__ float lds[8];
    i1 = block_reduce(i1, lds);
    o1 = block_reduce(o1, lds);
    m1 = block_reduce(m1, lds);
    i2 = block_reduce(i2, lds);
    v2 = block_reduce(v2, lds);
    m2 = block_reduce(m2, lds);
    if (threadIdx.x == 0) {
        const float u1 = o1 + m1 - i1;
        const float loss1 = 1.0f - (i1 + IOU_EPS) / (u1 + IOU_EPS);
        const float u2 = v2 + m2 - i2;
        const float loss2 = 1.0f - (i2 + IOU_EPS) / (u2 + IOU_EPS);
        out[0] = loss1;
        out[1] = loss2;
        out[2] = loss1 + loss2;   // A_COEF = B_COEF = 1
    }
}

// ---------------------------------------------------------------------------
// Host entry
// ---------------------------------------------------------------------------
extern "C" void kernel_launch(void* const* d_in, const int* in_sizes, int n_in,
                              void* d_out, int out_size, void* d_ws, size_t ws_size,
                              hipStream_t stream) {
    (void)n_in; (void)out_size; (void)ws_size;

    const float* o2d    = (const float*)d_in[0];   // output_2D  [512*8192]
    const float* m2d    = (const float*)d_in[1];   // mask_2D    [512*8192]
    const float* m3d    = (const float*)d_in[2];   // mask_3D    [D*D*D]
    const int*   index  = (const int*)  d_in[3];   // [N]
    const int*   midxyz = (const int*)  d_in[4];   // [N][3]

    const int n2d = in_sizes[0];                   // 4,194,304
    const int n3d = in_sizes[2];                   // 16,777,216
    const int N   = in_sizes[3];                   // 4,194,304
    const int D   = (int)llroundf(cbrtf((float)n3d)); // 256

    // Workspace: [ volume : n3d floats | partials : 3*(B2+B3) floats ]
    float* vol      = (float*)d_ws;
    float* partials = vol + n3d;

    const int T  = 256;       // 8 wave32s per block
    const int B2 = 512;       // blocks for 2D reduce
    const int B3 = 1024;      // blocks for 3D reduce

    k_zero<<<1024, T, 0, stream>>>((float4*)vol, n3d / 4);
    k_scatter<<<(N + T - 1) / T, T, 0, stream>>>(o2d, index, midxyz, vol, N, D);
    k_reduce<<<B2, T, 0, stream>>>((const float4*)o2d, (const float4*)m2d,
                                   n2d / 4, partials);
    k_reduce<<<B3, T, 0, stream>>>((const float4*)vol, (const float4*)m3d,
                                   n3d / 4, partials + 3 * (size_t)B2);
    k_final<<<1, T, 0, stream>>>(partials, B2, B3, (float*)d_out);
}